// CollectiveDecActorTaxi0Obs_33328946217203
// MI455X (gfx1250) — compile-verified
//
#include <hip/hip_runtime.h>
#include <hip/hip_bf16.h>

// ---- problem constants (fixed by the reference) ----
#define GRIDW     9
#define N_ZONES   81          // 9*9
#define OBS_DIM   48
#define MAX_ADJ   5
#define NCOL      405         // N_ZONES * MAX_ADJ
#define NTILES    26          // ceil(405/16), padded to 416
#define KSTEPS    12          // 48 / 4
#define ROWS_PER_WAVE 16
#define WAVES_PER_WG  2
#define LDS_STRIDE 420        // 416 cols + pad (420 mod 64 = 36 -> hi/lo halves hit different banks)
#define OUT_PER_ZONE 81
#define OUT_PER_ROW  6561     // 81*81
#define WAVE_OUT_ELEMS (ROWS_PER_WAVE * OUT_PER_ROW)   // 104976
#define WAVE_OUT_VEC4  (WAVE_OUT_ELEMS / 4)            // 26244
#define SCATTER_TASKS  (ROWS_PER_WAVE * NCOL)          // 6480

typedef __attribute__((ext_vector_type(2))) float v2f;
typedef __attribute__((ext_vector_type(4))) float v4f;
typedef __attribute__((ext_vector_type(8))) float v8f;

__global__ __launch_bounds__(WAVES_PER_WG * 32)
void CollectiveDecActorTaxi0Obs_33328946217203_kernel(
    const float* __restrict__ obs,   // [B, 48]
    const float* __restrict__ W,     // [81, 48, 5]
    const float* __restrict__ bias,  // [81, 5]
    const int*   __restrict__ idx,   // [81, 5]
    float* __restrict__ out)         // [B, 81, 81]
{
    __shared__ float smem[WAVES_PER_WG * ROWS_PER_WAVE * LDS_STRIDE];

    const int lane = threadIdx.x & 31;
    const int wave = threadIdx.x >> 5;
    const int b0   = (blockIdx.x * WAVES_PER_WG + wave) * ROWS_PER_WAVE;

    float* lds = smem + wave * (ROWS_PER_WAVE * LDS_STRIDE);
    float* outb = out + (size_t)b0 * OUT_PER_ROW;    // 16B aligned (b0 % 16 == 0)

    // ---------------- Stage 0: stream the wave's output slab to zero (NT b128 stores) --------
    // Issued first so the store stream overlaps the WMMA stage. Sparse scatter (stage 3)
    // follows in program order; same-wave stores to the same address complete in order.
    {
        const v4f z4 = {0.f, 0.f, 0.f, 0.f};
        v4f* p = (v4f*)outb + lane;
#pragma unroll 4
        for (int v = lane; v < WAVE_OUT_VEC4; v += 32) {
            __builtin_nontemporal_store(z4, p);
            p += 32;
        }
    }

    // ---------------- Stage 1: logits = obs_tile(16x48) x Wf(48x405) via WMMA f32 16x16x4 ----
    // A fragment (32-bit A 16x4): lanes 0-15 -> M=lane, K={0,1}; lanes 16-31 -> M=lane-16, K={2,3}
    const int mrow = lane & 15;
    const int kk   = (lane >> 4) << 1;              // 0 for lo half, 2 for hi half

    const float* arow = obs + (size_t)(b0 + mrow) * OBS_DIM + kk;
    v2f afrag[KSTEPS];
#pragma unroll
    for (int ks = 0; ks < KSTEPS; ++ks)
        afrag[ks] = *(const v2f*)(arow + ks * 4);   // K = 4*ks + kk, 4*ks + kk + 1

    for (int t = 0; t < NTILES; ++t) {
        // B fragment: lanes 0-15 -> K={k0,k0+1}, N=lane; lanes 16-31 -> K={k0+2,k0+3}, N=lane-16
        const int c  = t * 16 + mrow;               // logical column (zone*5 + adj)
        const int cc = (c < NCOL) ? c : (NCOL - 1); // clamp: pad columns compute garbage, never read
        const int zi = cc / MAX_ADJ;
        const int ji = cc - zi * MAX_ADJ;
        // Wf[d][c] = W[zi*240 + d*5 + ji]; row d = 4*ks + kk (+1)
        const float* wb = W + zi * (OBS_DIM * MAX_ADJ) + kk * MAX_ADJ + ji;

        v8f acc = {0.f, 0.f, 0.f, 0.f, 0.f, 0.f, 0.f, 0.f};
#pragma unroll
        for (int ks = 0; ks < KSTEPS; ++ks) {
            v2f bfrag;
            bfrag.x = wb[ks * 20];                  // d = 4*ks + kk
            bfrag.y = wb[ks * 20 + 5];              // d = 4*ks + kk + 1
            acc = __builtin_amdgcn_wmma_f32_16x16x4_f32(
                false, afrag[ks], false, bfrag, (short)0, acc, false, false);
        }

        // C/D layout: VGPR r -> M=r (lanes 0-15) / M=r+8 (lanes 16-31), N = lane&15
        const int col   = t * 16 + mrow;
        const int rbase = (lane >> 4) << 3;
#pragma unroll
        for (int r = 0; r < 8; ++r)
            lds[(rbase + r) * LDS_STRIDE + col] = acc[r];
    }

    __syncthreads();

    // ---------------- Stage 2: masked softmax over each 5-group, probs written back to LDS ---
    // adjacency slot order: [self, up(r>0), down(r<8), left(c>0), right(c<8)]
    for (int it = 0; it < (ROWS_PER_WAVE * N_ZONES + 31) / 32; ++it) {
        int t = it * 32 + lane;
        if (t < ROWS_PER_WAVE * N_ZONES) {
            int row = t / N_ZONES;
            int z   = t - row * N_ZONES;
            int r   = z / GRIDW;
            int cz  = z - r * GRIDW;
            int nv  = 1 + (r > 0) + (r < GRIDW - 1) + (cz > 0) + (cz < GRIDW - 1);

            float* p = lds + row * LDS_STRIDE + z * MAX_ADJ;
            float l[MAX_ADJ];
            float mx = -3.0e38f;
#pragma unroll
            for (int j = 0; j < MAX_ADJ; ++j) {
                l[j] = p[j] + bias[z * MAX_ADJ + j];
                if (j < nv) mx = fmaxf(mx, l[j]);
            }
            float s = 0.f;
#pragma unroll
            for (int j = 0; j < MAX_ADJ; ++j) {
                float e = (j < nv) ? __expf(l[j] - mx) : 0.f;
                l[j] = e;
                s += e;
            }
            float inv = __frcp_rn(s);
#pragma unroll
            for (int j = 0; j < MAX_ADJ; ++j)
                p[j] = l[j] * inv;
        }
    }

    __syncthreads();

    // ---------------- Stage 3: sparse scatter of the 16x405 probs into the zeroed slab -------
    for (int it = 0; it < (SCATTER_TASKS + 31) / 32; ++it) {
        int t = it * 32 + lane;
        if (t < SCATTER_TASKS) {
            int row = t / NCOL;
            int c   = t - row * NCOL;
            int z   = c / MAX_ADJ;
            int j   = c - z * MAX_ADJ;

            int r   = z / GRIDW;
            int cz  = z - r * GRIDW;
            int nv  = 1 + (r > 0) + (r < GRIDW - 1) + (cz > 0) + (cz < GRIDW - 1);

            if (j < nv) {   // predicated: padded slots must not clobber column idx==0
                int col  = idx[z * MAX_ADJ + j];
                float pv = lds[row * LDS_STRIDE + z * MAX_ADJ + j];
                outb[(size_t)row * OUT_PER_ROW + z * OUT_PER_ZONE + col] = pv;
            }
        }
    }
}

extern "C" void kernel_launch(void* const* d_in, const int* in_sizes, int n_in,
                              void* d_out, int out_size, void* d_ws, size_t ws_size,
                              hipStream_t stream) {
    const float* obs  = (const float*)d_in[0];
    const float* W    = (const float*)d_in[1];
    const float* bias = (const float*)d_in[2];
    const int*   idx  = (const int*)d_in[3];
    // d_in[4] = mask: validity derived analytically from the 9x9 grid.
    float* out = (float*)d_out;

    const int B = in_sizes[0] / OBS_DIM;                       // 32768
    const int nblocks = B / (ROWS_PER_WAVE * WAVES_PER_WG);    // 1024

    CollectiveDecActorTaxi0Obs_33328946217203_kernel
        <<<nblocks, WAVES_PER_WG * 32, 0, stream>>>(obs, W, bias, idx, out);
}